// ColAttention_2757369004753
// MI455X (gfx1250) — compile-verified
//
#include <hip/hip_runtime.h>

typedef __attribute__((ext_vector_type(2))) float v2f;
typedef __attribute__((ext_vector_type(8))) float v8f;

#define C_DIM 256
#define QK_D  64
#define H_DIM 96
#define W_DIM 96
#define B_DIM 16

// LDS leading dims, all == 4 (mod 32): 16-lane fragment accesses with the
// K-pair split (lanes 0-15 -> K0,K1; lanes 16-31 -> K2,K3) touch all 64 banks
// exactly once -> conflict-free ds_load_b64 everywhere.
#define XT_LD 260   // Xt / Yt : [h][c], c contiguous
#define XS_LD 100   // Xs      : [c][h], h contiguous
#define QT_LD 68    // Qt / Kt : [h][q], q contiguous
#define SS_LD 100   // Ss      : [h][g], g contiguous

// One fragment loader serves both operands:
//  A-fragment: element (m0+r, k)  at base[(m0+r)*ld + k]   (k contiguous)
//  B-fragment: element (k, n0+r)  at base[(n0+r)*ld + k]   (n-major, k contiguous)
__device__ __forceinline__ v2f frag(const float* __restrict__ base,
                                    int i0, int k0, int ld, int lane) {
  const int r  = lane & 15;
  const int kk = (lane >> 4) << 1;
  return *(const v2f*)(base + (i0 + r) * ld + (k0 + kk));
}

__device__ __forceinline__ v8f wmma_f32(v2f a, v2f b, v8f c) {
  return __builtin_amdgcn_wmma_f32_16x16x4_f32(
      false, a, false, b, (short)0, c, false, false);
}

extern "C" __global__ void __launch_bounds__(256)
col_attn_fused(const float* __restrict__ x,
               const float* __restrict__ Wq, const float* __restrict__ bq,
               const float* __restrict__ Wk, const float* __restrict__ bk,
               const float* __restrict__ Wv, const float* __restrict__ bv,
               const float* __restrict__ gamma,
               float* __restrict__ out) {
  extern __shared__ float smem[];
  float* Xt = smem;                       // 96 x 260  X^T (later reused as Y^T)
  float* Xs = Xt + H_DIM * XT_LD;         // 256 x 100 X
  float* Qt = Xs + C_DIM * XS_LD;         // 96 x 68   Q^T  [h][q]
  float* Kt = Qt + H_DIM * QT_LD;         // 96 x 68   K^T  [g][q]
  float* Ss = Kt + H_DIM * QT_LD;         // 96 x 100  scores/attn [h][g]
  float* Yt = Xt;                         // 96 x 260  Y^T  [h][c] (over dead Xt)

  const int blk  = blockIdx.x;
  const int b    = blk / W_DIM;
  const int w    = blk - b * W_DIM;
  const int tid  = threadIdx.x;
  const int lane = tid & 31;
  const int wave = tid >> 5;
  const int ncol = lane & 15;
  const int mo   = (lane >> 4) << 3;      // D-matrix: VGPR j holds M = m0+mo+j

  // ---- Phase 0: load X column into LDS twice (both majors) -----------------
  const float* xb = x + ((size_t)b * C_DIM * H_DIM) * W_DIM + w;
  for (int i = tid; i < C_DIM * H_DIM; i += 256) {
    const int c = i / H_DIM;
    const int h = i - c * H_DIM;
    const float v = xb[(size_t)(c * H_DIM + h) * W_DIM];
    Xs[c * XS_LD + h] = v;
    Xt[h * XT_LD + c] = v;
  }
  __syncthreads();

  // ---- Phase 1: Q^T = X^T Wq^T + bq, K^T = X^T Wk^T + bk  ([h][q], 96x64) --
  // 48 tiles = 8 blocks of (3 h-tiles x 2 q-tiles): exactly one block per wave.
  // Per k-step: 3 LDS A-frags + 2 global B-frags feed 6 WMMAs.
  {
    const int mat = wave >> 2;                  // 0 = Q, 1 = K
    const int mtb = ((wave >> 1) & 1) * 48;     // h tiles {0,16,32} or {48,64,80}
    const int nq  = (wave & 1) * 32;            // q tiles {0,16} or {32,48}
    const float* Wp = mat ? Wk : Wq;
    const float* bp = mat ? bk : bq;
    float*      Dst = mat ? Kt : Qt;

    v8f zero = {};
    v8f acc[3][2];
    for (int i = 0; i < 3; ++i)
      for (int jq = 0; jq < 2; ++jq) acc[i][jq] = zero;

    for (int k = 0; k < C_DIM; k += 4) {
      v2f a0 = frag(Xt, mtb,      k, XT_LD, lane);
      v2f a1 = frag(Xt, mtb + 16, k, XT_LD, lane);
      v2f a2 = frag(Xt, mtb + 32, k, XT_LD, lane);
      v2f b0 = frag(Wp, nq,      k, C_DIM, lane);   // B[c][q] = W[q][c]
      v2f b1 = frag(Wp, nq + 16, k, C_DIM, lane);
      acc[0][0] = wmma_f32(a0, b0, acc[0][0]);
      acc[0][1] = wmma_f32(a0, b1, acc[0][1]);
      acc[1][0] = wmma_f32(a1, b0, acc[1][0]);
      acc[1][1] = wmma_f32(a1, b1, acc[1][1]);
      acc[2][0] = wmma_f32(a2, b0, acc[2][0]);
      acc[2][1] = wmma_f32(a2, b1, acc[2][1]);
    }
    const float bb0 = bp[nq + ncol];
    const float bb1 = bp[nq + 16 + ncol];
    for (int i = 0; i < 3; ++i)
      for (int j = 0; j < 8; ++j) {
        const int row = mtb + i * 16 + mo + j;  // h
        Dst[row * QT_LD + nq + ncol]      = acc[i][0][j] + bb0;
        Dst[row * QT_LD + nq + 16 + ncol] = acc[i][1][j] + bb1;
      }
  }
  __syncthreads();

  // ---- Phase 2: S[h][g] = sum_q Qt[h][q] * Kt[g][q]  (96x96, K=64) ---------
  for (int t = wave; t < 36; t += 8) {
    const int mt = (t / 6) * 16;
    const int nt = (t % 6) * 16;
    v8f acc = {};
    for (int k = 0; k < QK_D; k += 4) {
      v2f a = frag(Qt, mt, k, QT_LD, lane);     // A[h][q]
      v2f bf = frag(Kt, nt, k, QT_LD, lane);    // B[q][g] = Kt[g][q]
      acc = wmma_f32(a, bf, acc);
    }
    for (int j = 0; j < 8; ++j)
      Ss[(mt + mo + j) * SS_LD + nt + ncol] = acc[j];
  }
  __syncthreads();

  // ---- Phase 3: softmax over g (one row per thread) ------------------------
  for (int r = tid; r < H_DIM; r += 256) {
    float* row = Ss + r * SS_LD;
    float m = -3.402823466e+38f;
    for (int g = 0; g < H_DIM; ++g) m = fmaxf(m, row[g]);
    float s = 0.f;
    for (int g = 0; g < H_DIM; ++g) { float e = __expf(row[g] - m); row[g] = e; s += e; }
    const float inv = 1.f / s;
    for (int g = 0; g < H_DIM; ++g) row[g] *= inv;
  }
  __syncthreads();

  // ---- Phase 4: Y^T[h][c] = sum_g A[h][g] * X[c][g]  (M=h, N=c, K=g) -------
  // A = attn rows of Ss (contig-K), B = Xs[c][g] (n-major, contig-K). D tiles
  // store straight into Yt[h][c] (contiguous, conflict-free). Yt overwrites
  // Xt, which is dead after phase 1. 16 blocks of (3 h-tiles x 2 c-tiles).
  for (int bk4 = wave; bk4 < 16; bk4 += 8) {
    const int mtb = (bk4 >> 3) * 48;            // h tile base
    const int nc  = (bk4 & 7) * 32;             // c tile pair base
    v8f zero = {};
    v8f acc[3][2];
    for (int i = 0; i < 3; ++i)
      for (int jn = 0; jn < 2; ++jn) acc[i][jn] = zero;

    for (int k = 0; k < H_DIM; k += 4) {        // k = g
      v2f a0 = frag(Ss, mtb,      k, SS_LD, lane);
      v2f a1 = frag(Ss, mtb + 16, k, SS_LD, lane);
      v2f a2 = frag(Ss, mtb + 32, k, SS_LD, lane);
      v2f b0 = frag(Xs, nc,      k, XS_LD, lane);  // B[g][c] = Xs[c][g]
      v2f b1 = frag(Xs, nc + 16, k, XS_LD, lane);
      acc[0][0] = wmma_f32(a0, b0, acc[0][0]);
      acc[0][1] = wmma_f32(a0, b1, acc[0][1]);
      acc[1][0] = wmma_f32(a1, b0, acc[1][0]);
      acc[1][1] = wmma_f32(a1, b1, acc[1][1]);
      acc[2][0] = wmma_f32(a2, b0, acc[2][0]);
      acc[2][1] = wmma_f32(a2, b1, acc[2][1]);
    }
    for (int i = 0; i < 3; ++i)
      for (int j = 0; j < 8; ++j) {
        const int row = mtb + i * 16 + mo + j;  // h
        Yt[row * XT_LD + nc + ncol]      = acc[i][0][j];
        Yt[row * XT_LD + nc + 16 + ncol] = acc[i][1][j];
      }
  }
  __syncthreads();

  // ---- Phase 5: Out^T[h][c_out] = sum_cin Yt[h][cin] * Wv[c_out][cin] ------
  // A = Yt rows (LDS contig-K), B = Wv rows (global b64, n-major contig-K).
  const float g0 = gamma[0];
  for (int bk5 = wave; bk5 < 16; bk5 += 8) {
    const int mtb = (bk5 >> 3) * 48;            // h tile base
    const int nco = (bk5 & 7) * 32;             // c_out tile pair base
    v8f zero = {};
    v8f acc[3][2];
    for (int i = 0; i < 3; ++i)
      for (int jn = 0; jn < 2; ++jn) acc[i][jn] = zero;

    for (int k = 0; k < C_DIM; k += 4) {        // k = c_in
      v2f a0 = frag(Yt, mtb,      k, XT_LD, lane);
      v2f a1 = frag(Yt, mtb + 16, k, XT_LD, lane);
      v2f a2 = frag(Yt, mtb + 32, k, XT_LD, lane);
      v2f b0 = frag(Wv, nco,      k, C_DIM, lane); // B[cin][cout] = Wv[cout][cin]
      v2f b1 = frag(Wv, nco + 16, k, C_DIM, lane);
      acc[0][0] = wmma_f32(a0, b0, acc[0][0]);
      acc[0][1] = wmma_f32(a0, b1, acc[0][1]);
      acc[1][0] = wmma_f32(a1, b0, acc[1][0]);
      acc[1][1] = wmma_f32(a1, b1, acc[1][1]);
      acc[2][0] = wmma_f32(a2, b0, acc[2][0]);
      acc[2][1] = wmma_f32(a2, b1, acc[2][1]);
    }
    const float bv0 = bv[nco + ncol];
    const float bv1 = bv[nco + 16 + ncol];
    for (int i = 0; i < 3; ++i)
      for (int j = 0; j < 8; ++j) {
        const int h  = mtb + i * 16 + mo + j;
        const int c0 = nco + ncol;
        const int c1 = nco + 16 + ncol;
        const size_t i0 = (((size_t)b * C_DIM + c0) * H_DIM + h) * W_DIM + w;
        const size_t i1 = (((size_t)b * C_DIM + c1) * H_DIM + h) * W_DIM + w;
        out[i0] = g0 * (acc[i][0][j] + bv0) + x[i0];
        out[i1] = g0 * (acc[i][1][j] + bv1) + x[i1];
      }
  }
}

extern "C" void kernel_launch(void* const* d_in, const int* in_sizes, int n_in,
                              void* d_out, int out_size, void* d_ws, size_t ws_size,
                              hipStream_t stream) {
  (void)in_sizes; (void)n_in; (void)out_size; (void)d_ws; (void)ws_size;
  const float* x     = (const float*)d_in[0];
  const float* Wq    = (const float*)d_in[1];
  const float* bq    = (const float*)d_in[2];
  const float* Wk    = (const float*)d_in[3];
  const float* bk    = (const float*)d_in[4];
  const float* Wv    = (const float*)d_in[5];
  const float* bv    = (const float*)d_in[6];
  const float* gamma = (const float*)d_in[7];
  float* out = (float*)d_out;

  const size_t shmem_bytes =
      (size_t)(H_DIM * XT_LD + C_DIM * XS_LD + 2 * H_DIM * QT_LD + H_DIM * SS_LD)
      * sizeof(float);                                   // 292,864 B <= 320 KB

  (void)hipFuncSetAttribute(reinterpret_cast<const void*>(col_attn_fused),
                            hipFuncAttributeMaxDynamicSharedMemorySize,
                            (int)shmem_bytes);

  col_attn_fused<<<dim3(B_DIM * W_DIM), dim3(256), shmem_bytes, stream>>>(
      x, Wq, bq, Wk, bk, Wv, bv, gamma, out);
}